// InvGridSamplerDecomposed_3066606649875
// MI455X (gfx1250) — compile-verified
//
#include <hip/hip_runtime.h>
#include <hip/hip_bf16.h>
#include <stdint.h>

// ---------------------------------------------------------------------------
// InvGridSamplerDecomposed on MI455X (gfx1250)
//
// Pass 1: zero workspace A (B,C,515,515) + D (B,515,515)           [~72 MB]
// Pass 2: scatter — block = 256 pixels of one row (b,i). TDM tensor_load_to_lds
//         stages x[b, 0:16, i, j0:j0+256] (16 KB, 2D tile: tile 256x16,
//         tensor_dim0_stride = H*W) into LDS; s_wait_tensorcnt 0; then each
//         thread does 4 D atomics + 16x4 A atomics (global_atomic_add_f32,
//         non-returning, STOREcnt only).
// Pass 3: normalize — per wave: one 16ch x 16px tile,
//         out = A x diag(r) + H via 4x V_WMMA_F32_16X16X4_F32 (f32-exact),
//         r = 1/(D+eps) (valid) else 0, H = hole-fill broadcast.
// ---------------------------------------------------------------------------

#define BB 4
#define CC 16
#define HH 512
#define WW 512
#define H3 (HH + 3)   // 515
#define W3 (WW + 3)   // 515
#define EPSF 1e-10f

#define TILE_W 256    // scatter block pixel tile (divides WW)

#if defined(__HIP_DEVICE_COMPILE__) && __has_builtin(__builtin_amdgcn_tensor_load_to_lds) && __has_builtin(__builtin_amdgcn_s_wait_tensorcnt)
#define USE_TDM 1
#else
#define USE_TDM 0
#endif

#if defined(__HIP_DEVICE_COMPILE__) && __has_builtin(__builtin_amdgcn_wmma_f32_16x16x4_f32)
#define USE_WMMA 1
#else
#define USE_WMMA 0
#endif

typedef float    v2f  __attribute__((ext_vector_type(2)));
typedef float    v8f  __attribute__((ext_vector_type(8)));
typedef unsigned v4u  __attribute__((ext_vector_type(4)));
typedef int      v4i_ __attribute__((ext_vector_type(4)));
typedef int      v8i_ __attribute__((ext_vector_type(8)));

__global__ void igs_zero_kernel(float* __restrict__ p, int n) {
    int i = blockIdx.x * blockDim.x + threadIdx.x;
    if (i < n) p[i] = 0.0f;
}

// ---------------------------------------------------------------------------
// Pass 2: scatter
// ---------------------------------------------------------------------------
__global__ void igs_scatter_kernel(const float* __restrict__ x,
                                   const float* __restrict__ grid,
                                   float* __restrict__ A,
                                   float* __restrict__ D) {
    __shared__ float xs[CC * TILE_W];            // 16 KB channel-major tile

    const int t   = threadIdx.x;                 // 0..255 = pixel within tile
    const int pid = blockIdx.x * TILE_W + t;     // global source pixel
    const int j   = pid & (WW - 1);
    const int i   = (pid >> 9) & (HH - 1);
    const int b   = pid >> 18;                   // H*W = 2^18

#if USE_TDM
    // --- Tensor Data Mover: stage x[b, 0:CC, i, j0:j0+TILE_W] into LDS ------
    if (t < 32) {                                // wave 0 issues the DMA
        const int j0 = blockIdx.x * TILE_W & (WW - 1);
        const float* tile = x + (((size_t)b * CC) * HH + i) * WW + j0;
        uint64_t gaddr   = (uint64_t)(const void*)tile;         // byte address
        uint32_t lds_off = (uint32_t)(size_t)(void*)&xs[0];     // LDS byte offset

        // D# group 0: count=1 | lds_addr | global_addr[56:0] | type=2
        v4u g0;
        g0.x = 1u;
        g0.y = lds_off;
        g0.z = (uint32_t)(gaddr & 0xffffffffull);
        g0.w = (uint32_t)((gaddr >> 32) & 0x1ffffffull) | (2u << 30);

        // D# group 1: data_size=4B, tensor 256 x 16, tile 256 x 16,
        //             tensor_dim0_stride = H*W elements (channel stride)
        const uint32_t td0 = TILE_W, td1 = CC;     // tensor dims (no OOB)
        const uint32_t tl0 = TILE_W, tl1 = CC;     // tile dims
        const uint64_t s0  = (uint64_t)HH * WW;    // dim0 stride (elements)
        uint32_t w0 = (2u << 16);                                  // data_size=2 (4B)
        uint32_t w1 = (td0 & 0xffffu) << 16;                       // td0[15:0] @63:48
        uint32_t w2 = (td0 >> 16) | ((td1 & 0xffffu) << 16);       // td0[31:16], td1[15:0]
        uint32_t w3 = (td1 >> 16) | (tl0 << 16);                   // td1[31:16], tile0
        uint32_t w4 = tl1 & 0xffffu;                               // tile1; tile2=0
        uint32_t w5 = (uint32_t)(s0 & 0xffffffffull);              // stride0[31:0]
        uint32_t w6 = (uint32_t)((s0 >> 32) & 0xffffull);          // stride0[47:32]
        v8i_ g1 = { (int)w0, (int)w1, (int)w2, (int)w3,
                    (int)w4, (int)w5, (int)w6, 0 };
        v4i_ z4 = { 0, 0, 0, 0 };

#if __has_include(<hip/amd_detail/amd_gfx1250_TDM.h>)
        v8i_ z8 = { 0, 0, 0, 0, 0, 0, 0, 0 };     // amdgpu-toolchain: 6-arg form
        __builtin_amdgcn_tensor_load_to_lds(g0, g1, z4, z4, z8, 0);
#else
        __builtin_amdgcn_tensor_load_to_lds(g0, g1, z4, z4, 0);   // ROCm 7.2: 5-arg
#endif
        __builtin_amdgcn_s_wait_tensorcnt(0);    // issuing wave waits for TDM
    }
    __syncthreads();                             // publish LDS tile to all waves
#endif

    float2 gv = reinterpret_cast<const float2*>(grid)[pid];

    // g = (inv_grid+1)*0.5 ; gi = clip(g0*H + 1, 0, H+1-2eps) (f32 bound == 513.0f)
    float gi = fminf(fmaxf((gv.x + 1.0f) * 0.5f * (float)HH + 1.0f, 0.0f), 513.0f);
    float gj = fminf(fmaxf((gv.y + 1.0f) * 0.5f * (float)WW + 1.0f, 0.0f), 513.0f);

    int ii0 = (int)gi;                // gi >= 1.0 => trunc == floor
    int jj0 = (int)gj;
    float fi = gi - (float)ii0;
    float fj = gj - (float)jj0;

    float w00 = (1.0f - fi) * (1.0f - fj);
    float w01 = (1.0f - fi) * fj;
    float w10 = fi * (1.0f - fj);
    float w11 = fi * fj;

    // D scatter (once per pixel, amortized over the 16 channels)
    float* dp = D + ((size_t)b * H3 + ii0) * W3 + jj0;
    atomicAdd(dp,          w00);
    atomicAdd(dp + 1,      w01);
    atomicAdd(dp + W3,     w10);
    atomicAdd(dp + W3 + 1, w11);

    float* ap = A + (((size_t)b * CC) * H3 + ii0) * W3 + jj0;
    const size_t astride = (size_t)H3 * W3;

#if USE_TDM
#pragma unroll 4
    for (int c = 0; c < CC; ++c) {
        float xv = xs[c * TILE_W + t];           // ds_load_b32, bank-conflict-free
        atomicAdd(ap,          xv * w00);
        atomicAdd(ap + 1,      xv * w01);
        atomicAdd(ap + W3,     xv * w10);
        atomicAdd(ap + W3 + 1, xv * w11);
        ap += astride;
    }
#else
    const float* xp = x + (((size_t)b * CC) * HH + i) * WW + j;
    const size_t xstride = (size_t)HH * WW;
#pragma unroll 4
    for (int c = 0; c < CC; ++c) {
        __builtin_prefetch(xp + xstride, 0, 0);  // -> global_prefetch_b8
        float xv = *xp;
        atomicAdd(ap,          xv * w00);
        atomicAdd(ap + 1,      xv * w01);
        atomicAdd(ap + W3,     xv * w10);
        atomicAdd(ap + W3 + 1, xv * w11);
        xp += xstride;
        ap += astride;
    }
#endif
}

// ---------------------------------------------------------------------------
// Pass 3: normalize. One wave per (b, y, x0) tile: 16 channels x 16 pixels.
// Waves total = B*H*(W/16) = 65536; launched as 8192 blocks x 256 threads.
// ---------------------------------------------------------------------------
__global__ void igs_normalize_kernel(const float* __restrict__ A,
                                     const float* __restrict__ D,
                                     float* __restrict__ out) {
    const int lane   = threadIdx.x & 31;
    const int waveId = blockIdx.x * (blockDim.x >> 5) + (threadIdx.x >> 5);

    const int b    = waveId >> 14;               // H*(W/16) = 2^14 tiles per b
    const int rem  = waveId & 0x3fff;
    const int y    = rem >> 5;                   // W/16 = 32 tiles per row
    const int x0   = (rem & 31) << 4;

    const int  nB    = lane & 15;                // pixel column of this lane
    const bool hiHalf = lane >= 16;

    // Per-pixel scale r(n) and hole-fill h(n)
    float d = D[((size_t)b * H3 + (y + 1)) * W3 + (x0 + 1 + nB)];
    bool  valid = d > EPSF;
    float r = valid ? (1.0f / (d + EPSF)) : 0.0f;
    float h = valid ? 0.0f : 1.0f;

#if USE_WMMA
    // C = H (hole fill broadcast over channels): C(m,n) = h(n)
    v8f acc;
#pragma unroll
    for (int v = 0; v < 8; ++v) acc[v] = h;

    // A-matrix (16x16 data, §7.12.2 layout) x B = diag(r), 4 chunks of K=4.
    const int kA = hiHalf ? 2 : 0;               // A/B: VGPR0 holds K = kA, VGPR1 K = kA+1
    const float* arow =
        A + (((size_t)b * CC + nB) * H3 + (y + 1)) * W3 + (x0 + 1);  // row c = lane&15

#pragma unroll
    for (int kb = 0; kb < 4; ++kb) {
        v2f a, bm;
        a.x = arow[4 * kb + kA];                 // A(c=lane&15, p=4kb+kA)
        a.y = arow[4 * kb + kA + 1];
        bm.x = (nB == 4 * kb + kA)     ? r : 0.0f;   // diag(r) chunk
        bm.y = (nB == 4 * kb + kA + 1) ? r : 0.0f;
        acc = __builtin_amdgcn_wmma_f32_16x16x4_f32(
                  false, a, false, bm, (short)0, acc, false, false);
    }

    // D-matrix store: VGPR v -> channel m = v (+8 for lanes 16-31), col = nB.
    // Two contiguous 64B segments per store -> fully coalesced.
#pragma unroll
    for (int v = 0; v < 8; ++v) {
        int m = v + (hiHalf ? 8 : 0);
        out[(((size_t)b * CC + m) * HH + y) * WW + (x0 + nB)] = acc[v];
    }
#else
    // Scalar fallback with the identical tile contract (8 elements per lane).
#pragma unroll
    for (int v = 0; v < 8; ++v) {
        int m = v + (hiHalf ? 8 : 0);
        float a = A[(((size_t)b * CC + m) * H3 + (y + 1)) * W3 + (x0 + 1 + nB)];
        out[(((size_t)b * CC + m) * HH + y) * WW + (x0 + nB)] =
            valid ? (a * r + h) : 1.0f;
    }
#endif
}

extern "C" void kernel_launch(void* const* d_in, const int* in_sizes, int n_in,
                              void* d_out, int out_size, void* d_ws, size_t ws_size,
                              hipStream_t stream) {
    const float* x    = (const float*)d_in[0];   // (4,16,512,512) f32
    const float* grid = (const float*)d_in[1];   // (4,512,512,2)  f32
    float* out = (float*)d_out;                  // (4,16,512,512) f32

    const size_t ACOUNT = (size_t)BB * CC * H3 * W3;   // 16,974,400 floats
    const size_t DCOUNT = (size_t)BB * H3 * W3;        //  1,060,900 floats
    float* A = (float*)d_ws;
    float* D = A + ACOUNT;

    const int zn = (int)(ACOUNT + DCOUNT);
    igs_zero_kernel<<<(zn + 255) / 256, 256, 0, stream>>>(A, zn);

    const int npix = BB * HH * WW;               // 1,048,576
    igs_scatter_kernel<<<npix / TILE_W, TILE_W, 0, stream>>>(x, grid, A, D);

    const int ntiles = BB * HH * (WW / 16);      // 65,536 waves
    igs_normalize_kernel<<<ntiles / 8, 256, 0, stream>>>(A, D, out);
}